// DummyConv_2439541424746
// MI455X (gfx1250) — compile-verified
//
#include <hip/hip_runtime.h>
#include <hip/hip_bf16.h>

typedef __attribute__((ext_vector_type(2))) float v2f;
typedef __attribute__((ext_vector_type(8))) float v8f;

#define HW   8192   // data is HW x HW
#define GRID 512    // output grid is GRID x GRID
#define E    256

// ---------------------------------------------------------------------------
// Kernel 1: reduce weight (E x E, row-major: [out][in]) over the OUT dim to
// wsum[in] and bias to a scalar bsum.  ws[0..255] = wsum, ws[256] = bsum.
// One block of 256 threads; 64K strided f32 reads — negligible cost.
// ---------------------------------------------------------------------------
__global__ void dummyconv_prep_kernel(const float* __restrict__ weight,
                                      const float* __restrict__ bias,
                                      float* __restrict__ ws) {
    const int j = threadIdx.x;            // 0..255 = input-feature index
    float s = 0.0f;
    #pragma unroll 8
    for (int i = 0; i < E; ++i) s += weight[i * E + j];
    ws[j] = s;

    __shared__ float red[E];
    red[j] = bias[j];
    __syncthreads();
    for (int off = E / 2; off > 0; off >>= 1) {
        if (j < off) red[j] += red[j + off];
        __syncthreads();
    }
    if (j == 0) ws[E] = red[0];
}

// ---------------------------------------------------------------------------
// Kernel 2: streaming pass over data.  One wave32 -> 16 adjacent patches
// (a 16-row x 1KB stripe).  Dot products done with V_WMMA_F32_16X16X4_F32:
//   A(16x4) = 16 patches x 4 features, B(4x16) = wsum chunk broadcast to all
//   columns, C accumulates over 64 WMMAs (K = 256).
// Feature->k mapping chosen so each lane's A data for two consecutive WMMAs
// is one contiguous float4 (global_load_b128), and likewise for wsum in LDS.
// ---------------------------------------------------------------------------
__global__ void __launch_bounds__(256)
dummyconv_main_kernel(const float* __restrict__ data,
                      const float* __restrict__ ws,
                      float* __restrict__ out) {
    __shared__ float lws[E + 1];
    const int tid = threadIdx.x;          // 256 threads = 8 waves
    lws[tid] = ws[tid];
    if (tid == 0) lws[E] = ws[E];
    __syncthreads();

    const int wave = tid >> 5;
    const int lane = tid & 31;
    const int task = blockIdx.x * 8 + wave;       // 0..16383
    const int gy   = task >> 5;                   // patch-grid row   0..511
    const int gx0  = (task & 31) << 4;            // first patch col  0..496

    const int m = lane & 15;              // A-matrix row: patch index
    const int h = lane >> 4;              // k-half: 0 -> K={0,1}, 1 -> K={2,3}

    // Base of this lane's patch (patch = 16x16 block of row-major data).
    const float* patchBase =
        data + (size_t)gy * 16 * HW + (size_t)(gx0 + m) * 16;

    v8f acc = {};                         // C/D: 16x16 f32, 8 VGPRs

    #pragma unroll
    for (int p = 0; p < 32; ++p) {
        // Feature block for this lane: fb..fb+3 (contiguous within a patch row
        // since fb % 4 == 0 and rows are 16 wide).
        const int fb = p * 8 + h * 4;
        const int r  = fb >> 4;           // row inside patch
        const int c  = fb & 15;           // col inside patch

        const float4 a  = *(const float4*)(patchBase + (size_t)r * HW + c);
        const float4 wv = *(const float4*)(&lws[fb]);

        // WMMA #1: k-pair (a.x, a.y) against wsum pair (wv.x, wv.y)
        v2f A0 = {a.x, a.y};
        v2f B0 = {wv.x, wv.y};
        acc = __builtin_amdgcn_wmma_f32_16x16x4_f32(
                  false, A0, false, B0, (short)0, acc, false, false);

        // WMMA #2: k-pair (a.z, a.w) against (wv.z, wv.w)
        v2f A1 = {a.z, a.w};
        v2f B1 = {wv.z, wv.w};
        acc = __builtin_amdgcn_wmma_f32_16x16x4_f32(
                  false, A1, false, B1, (short)0, acc, false, false);
    }

    // D layout: lanes 0-15 VGPR v holds row M=v; lanes 16-31 hold M=v+8.
    // All 16 columns are identical (B columns replicated), so any lane in the
    // right half has patch M's value in VGPR (M & 7).
    // Lanes 0-7 write patches 0-7; lanes 16-23 write patches 8-15.
    const float bsum = lws[E];
    if ((lane & 8) == 0) {
        const int r7 = lane & 7;
        float v01 = (r7 & 1) ? acc[1] : acc[0];
        float v23 = (r7 & 1) ? acc[3] : acc[2];
        float v45 = (r7 & 1) ? acc[5] : acc[4];
        float v67 = (r7 & 1) ? acc[7] : acc[6];
        float v0123 = (r7 & 2) ? v23 : v01;
        float v4567 = (r7 & 2) ? v67 : v45;
        float v = (r7 & 4) ? v4567 : v0123;

        const int patch = h * 8 + r7;     // = M
        out[(size_t)gy * GRID + gx0 + patch] = v + bsum;
    }
}

extern "C" void kernel_launch(void* const* d_in, const int* in_sizes, int n_in,
                              void* d_out, int out_size, void* d_ws, size_t ws_size,
                              hipStream_t stream) {
    const float* data   = (const float*)d_in[0];  // 8192 x 8192 f32
    const float* weight = (const float*)d_in[1];  // 256 x 256 f32
    const float* bias   = (const float*)d_in[2];  // 256 f32
    float*       out    = (float*)d_out;          // 512 x 512 f32
    float*       ws     = (float*)d_ws;           // [0..255]=wsum, [256]=bsum

    dummyconv_prep_kernel<<<dim3(1), dim3(256), 0, stream>>>(weight, bias, ws);

    // 512*512 patches / 16 patches-per-wave = 16384 waves = 2048 blocks x 8 waves
    dummyconv_main_kernel<<<dim3(2048), dim3(256), 0, stream>>>(data, ws, out);
}